// GAT_56745107915455
// MI455X (gfx1250) — compile-verified
//
#include <hip/hip_runtime.h>
#include <hip/hip_bf16.h>
#include <math.h>

#define HEADS   2
#define HID     64
#define HC      128      // HEADS*HID == F_IN
#define NCLASS  10
#define NGRAPH  64
#define NEG_SLOPE 0.2f
#define LDS_STRIDE (HC + 4)   // 132 dwords: 16B-aligned rows, banks spread by 4
#define ROWS_PER_BLOCK 64     // 4 row-tiles of 16

typedef __attribute__((ext_vector_type(2))) float v2f;
typedef __attribute__((ext_vector_type(8))) float v8f;

// ---------------------------------------------------------------------------
// float atomic max via sign-aware int/uint trick (target initialized to -inf)
// ---------------------------------------------------------------------------
__device__ __forceinline__ void atomicMaxF(float* addr, float val) {
    if (val >= 0.0f) atomicMax((int*)addr, __float_as_int(val));
    else             atomicMin((unsigned int*)addr, __float_as_uint(val));
}

// ---------------------------------------------------------------------------
// store one 16x16 f32 C/D tile (8 VGPRs/lane) with a single bounds branch
// ---------------------------------------------------------------------------
__device__ __forceinline__ void store_ctile(float* __restrict__ H, v8f acc,
                                            int rowBase, int colOut, int n) {
    float* Hp = H + (size_t)rowBase * HC + colOut;
    if (rowBase + 7 < n) {
#pragma unroll
        for (int r = 0; r < 8; ++r) Hp[(size_t)r * HC] = acc[r];
    } else {
#pragma unroll
        for (int r = 0; r < 8; ++r)
            if (rowBase + r < n) Hp[(size_t)r * HC] = acc[r];
    }
}

// ---------------------------------------------------------------------------
// H = X @ W  with V_WMMA_F32_16X16X4_F32  (X: [n,128], W: [128,128] row-major)
// block = 256 threads = 8 waves; block owns a 64-row strip (4 row-tiles),
// each wave one 16-wide column tile with 4 accumulators -> each B fragment
// is reused 4x from registers.  A staged in LDS (coalesced b128 loads,
// conflict-free ds_load_b64 fragment reads).  128 WMMA issues per wave.
// ---------------------------------------------------------------------------
__global__ void gat_gemm_wmma(const float* __restrict__ X,
                              const float* __restrict__ W,
                              float* __restrict__ H, int n) {
    __shared__ float As[ROWS_PER_BLOCK * LDS_STRIDE];   // 64 x 132 f32 = 33 KB

    const int tid  = threadIdx.x;
    const int wave = tid >> 5;                 // 0..7 -> column tile
    const int lane = tid & 31;
    const int row0 = blockIdx.x * ROWS_PER_BLOCK;

    // cooperative, coalesced stage of the 64x128 A tile (2048 float4 slots)
    for (int v = tid; v < ROWS_PER_BLOCK * (HC / 4); v += 256) {
        int r  = v >> 5;                       // 32 float4 per row
        int cc = (v & 31) << 2;
        int rr = row0 + r; if (rr >= n) rr = n - 1;
        float4 val = *(const float4*)(X + (size_t)rr * HC + cc);
        *(float4*)&As[r * LDS_STRIDE + cc] = val;   // 16B aligned (528B rows)
    }
    __syncthreads();

    const int m     = lane & 15;
    const int khalf = (lane >> 4) << 1;        // 0 or 2 (K split across lane halves)
    const int col0  = wave << 4;
    const float* Bcol = W + col0 + m;
    const float* A0   = &As[m * LDS_STRIDE + khalf];

    v8f acc0 = {}, acc1 = {}, acc2 = {}, acc3 = {};
#pragma unroll 4
    for (int k = 0; k < HC; k += 4) {
        v2f b;
        b.x = Bcol[(size_t)(k + khalf)     * HC];
        b.y = Bcol[(size_t)(k + khalf + 1) * HC];
        v2f a0 = *(const v2f*)(A0 + k);
        v2f a1 = *(const v2f*)(A0 + 16 * LDS_STRIDE + k);
        v2f a2 = *(const v2f*)(A0 + 32 * LDS_STRIDE + k);
        v2f a3 = *(const v2f*)(A0 + 48 * LDS_STRIDE + k);
        acc0 = __builtin_amdgcn_wmma_f32_16x16x4_f32(false, a0, false, b, (short)0, acc0, false, false);
        acc1 = __builtin_amdgcn_wmma_f32_16x16x4_f32(false, a1, false, b, (short)0, acc1, false, false);
        acc2 = __builtin_amdgcn_wmma_f32_16x16x4_f32(false, a2, false, b, (short)0, acc2, false, false);
        acc3 = __builtin_amdgcn_wmma_f32_16x16x4_f32(false, a3, false, b, (short)0, acc3, false, false);
    }

    // C/D layout: lanes 0-15 -> M = r, N = lane; lanes 16-31 -> M = r+8
    const int colOut  = col0 + m;
    const int rowBase = row0 + ((lane >> 4) << 3);
    store_ctile(H, acc0, rowBase,      colOut, n);
    store_ctile(H, acc1, rowBase + 16, colOut, n);
    store_ctile(H, acc2, rowBase + 32, colOut, n);
    store_ctile(H, acc3, rowBase + 48, colOut, n);
}

// ---------------------------------------------------------------------------
// alpha_src / alpha_dst: per (node, head) dot product of 64 channels
// ---------------------------------------------------------------------------
__global__ void gat_alpha(const float* __restrict__ H,
                          const float* __restrict__ a_src,
                          const float* __restrict__ a_dst,
                          float* __restrict__ as, float* __restrict__ ad, int n) {
    int i = blockIdx.x * blockDim.x + threadIdx.x;
    if (i >= n * HEADS) return;
    int node = i >> 1, head = i & 1;
    const float4* hp  = (const float4*)(H + (size_t)node * HC + head * HID);
    const float4* asp = (const float4*)(a_src + head * HID);
    const float4* adp = (const float4*)(a_dst + head * HID);
    float s = 0.f, d = 0.f;
#pragma unroll 4
    for (int c = 0; c < HID / 4; ++c) {
        float4 v = hp[c], va = asp[c], vd = adp[c];
        s += v.x * va.x + v.y * va.y + v.z * va.z + v.w * va.w;
        d += v.x * vd.x + v.y * vd.y + v.z * vd.z + v.w * vd.w;
    }
    as[i] = s; ad[i] = d;
}

// ---------------------------------------------------------------------------
// per-layer init: zero message accumulator (float4), m = -inf, denom = 0
// ---------------------------------------------------------------------------
__global__ void gat_init(float* __restrict__ Out, float* __restrict__ m,
                         float* __restrict__ den, int n) {
    int i = blockIdx.x * blockDim.x + threadIdx.x;
    if (i < n * (HC / 4)) ((float4*)Out)[i] = make_float4(0.f, 0.f, 0.f, 0.f);
    if (i < n * HEADS) { m[i] = -INFINITY; den[i] = 0.f; }
}

// ---------------------------------------------------------------------------
// edge pass 1: e = leaky_relu(as[src]+ad[dst]); segment max over dst
// one thread per edge, both heads (b64 loads/stores)
// edges [0,E) from edge_index, [E, E+n) are self loops
// ---------------------------------------------------------------------------
__global__ void gat_edge_max(const int* __restrict__ srcI, const int* __restrict__ dstI,
                             int E, int n,
                             const float* __restrict__ as, const float* __restrict__ ad,
                             float* __restrict__ ev, float* __restrict__ m) {
    int e = blockIdx.x * blockDim.x + threadIdx.x;
    if (e >= E + n) return;
    int s, d;
    if (e < E) { s = srcI[e]; d = dstI[e]; } else { s = d = e - E; }
    float2 vs = *(const float2*)(as + (size_t)s * HEADS);
    float2 vd = *(const float2*)(ad + (size_t)d * HEADS);
    float v0 = vs.x + vd.x; v0 = (v0 > 0.f) ? v0 : NEG_SLOPE * v0;
    float v1 = vs.y + vd.y; v1 = (v1 > 0.f) ? v1 : NEG_SLOPE * v1;
    *(float2*)(ev + (size_t)e * HEADS) = make_float2(v0, v1);
    atomicMaxF(&m[d * HEADS + 0], v0);
    atomicMaxF(&m[d * HEADS + 1], v1);
}

// ---------------------------------------------------------------------------
// edge pass 2: e = exp(e - m[dst]); denom[dst] += e   (both heads per thread)
// ---------------------------------------------------------------------------
__global__ void gat_edge_exp(const int* __restrict__ dstI, int E, int n,
                             const float* __restrict__ m,
                             float* __restrict__ ev, float* __restrict__ den) {
    int e = blockIdx.x * blockDim.x + threadIdx.x;
    if (e >= E + n) return;
    int d = (e < E) ? dstI[e] : (e - E);
    float2 mm = *(const float2*)(m + (size_t)d * HEADS);
    float2 vv = *(const float2*)(ev + (size_t)e * HEADS);
    float v0 = __expf(vv.x - mm.x);
    float v1 = __expf(vv.y - mm.y);
    *(float2*)(ev + (size_t)e * HEADS) = make_float2(v0, v1);
    atomicAdd(&den[d * HEADS + 0], v0);
    atomicAdd(&den[d * HEADS + 1], v1);
}

// ---------------------------------------------------------------------------
// edge pass 3: Out[dst, c..c+3] += H[src, c..c+3] * (e / (denom[dst]+1e-16))
// one thread per (edge, 4 channels): 32 lanes cover one edge's 512B row ->
// fully coalesced global_load_b128 gathers.
// ---------------------------------------------------------------------------
__global__ void gat_edge_scatter(const int* __restrict__ srcI, const int* __restrict__ dstI,
                                 int E, int n,
                                 const float* __restrict__ H,
                                 const float* __restrict__ ev, const float* __restrict__ den,
                                 float* __restrict__ Out) {
    size_t i = (size_t)blockIdx.x * blockDim.x + threadIdx.x;
    size_t tot = (size_t)(E + n) * (HC / 4);
    if (i >= tot) return;
    int    c = (int)(i & (HC / 4 - 1)) << 2;   // 0,4,...,124
    size_t e = i >> 5;
    int head = c >> 6;
    int s, d;
    if (e < (size_t)E) { s = srcI[e]; d = dstI[e]; } else { s = d = (int)(e - E); }
    float alpha = ev[e * HEADS + head] / (den[d * HEADS + head] + 1e-16f);
    float4 hv = *(const float4*)(H + (size_t)s * HC + c);
    float* o = Out + (size_t)d * HC + c;
    atomicAdd(o + 0, hv.x * alpha);
    atomicAdd(o + 1, hv.y * alpha);
    atomicAdd(o + 2, hv.z * alpha);
    atomicAdd(o + 3, hv.w * alpha);
}

// ---------------------------------------------------------------------------
// bias (+ optional ELU) epilogue -> next layer input (float4)
// ---------------------------------------------------------------------------
__global__ void gat_epilogue(const float* __restrict__ Out, const float* __restrict__ b,
                             float* __restrict__ Xn, int n, int do_elu) {
    int i = blockIdx.x * blockDim.x + threadIdx.x;
    if (i >= n * (HC / 4)) return;
    int c = (i & (HC / 4 - 1)) << 2;
    float4 v  = ((const float4*)Out)[i];
    float4 bb = *(const float4*)(b + c);
    v.x += bb.x; v.y += bb.y; v.z += bb.z; v.w += bb.w;
    if (do_elu) {
        v.x = (v.x > 0.f) ? v.x : (__expf(v.x) - 1.f);
        v.y = (v.y > 0.f) ? v.y : (__expf(v.y) - 1.f);
        v.z = (v.z > 0.f) ? v.z : (__expf(v.z) - 1.f);
        v.w = (v.w > 0.f) ? v.w : (__expf(v.w) - 1.f);
    }
    ((float4*)Xn)[i] = v;
}

// ---------------------------------------------------------------------------
// pooling
// ---------------------------------------------------------------------------
__global__ void pool_init(float* __restrict__ pooled, float* __restrict__ cnts) {
    int i = blockIdx.x * blockDim.x + threadIdx.x;
    if (i < NGRAPH * HC) pooled[i] = 0.f;
    if (i < NGRAPH) cnts[i] = 0.f;
}

__global__ void pool_sum(const float* __restrict__ X, const int* __restrict__ batch,
                         float* __restrict__ pooled, float* __restrict__ cnts, int n) {
    int i = blockIdx.x * blockDim.x + threadIdx.x;
    if (i >= n * (HC / 4)) return;
    int node = i >> 5, c = (i & (HC / 4 - 1)) << 2;
    int g = batch[node];
    float4 v = ((const float4*)X)[i];
    float* p = pooled + g * HC + c;
    atomicAdd(p + 0, v.x);
    atomicAdd(p + 1, v.y);
    atomicAdd(p + 2, v.z);
    atomicAdd(p + 3, v.w);
    if (c == 0) atomicAdd(&cnts[g], 1.f);
}

// ---------------------------------------------------------------------------
// final linear + softmax: one thread per graph
// ---------------------------------------------------------------------------
__global__ void final_head(const float* __restrict__ pooled, const float* __restrict__ cnts,
                           const float* __restrict__ lw, const float* __restrict__ lb,
                           float* __restrict__ out) {
    int g = threadIdx.x;
    if (g >= NGRAPH) return;
    float inv = 1.f / fmaxf(cnts[g], 1.f);
    float logits[NCLASS];
#pragma unroll
    for (int k = 0; k < NCLASS; ++k) logits[k] = lb[k];
    for (int c = 0; c < HC; ++c) {
        float p = pooled[g * HC + c] * inv;
#pragma unroll
        for (int k = 0; k < NCLASS; ++k) logits[k] += p * lw[c * NCLASS + k];
    }
    float mx = logits[0];
#pragma unroll
    for (int k = 1; k < NCLASS; ++k) mx = fmaxf(mx, logits[k]);
    float ssum = 0.f;
#pragma unroll
    for (int k = 0; k < NCLASS; ++k) { logits[k] = __expf(logits[k] - mx); ssum += logits[k]; }
#pragma unroll
    for (int k = 0; k < NCLASS; ++k) out[g * NCLASS + k] = logits[k] / ssum;
}

// ---------------------------------------------------------------------------
extern "C" void kernel_launch(void* const* d_in, const int* in_sizes, int n_in,
                              void* d_out, int out_size, void* d_ws, size_t ws_size,
                              hipStream_t stream) {
    const float* x     = (const float*)d_in[0];
    const int*   ei    = (const int*)d_in[1];
    const int*   batch = (const int*)d_in[2];
    const float* Wl[3]   = {(const float*)d_in[3],  (const float*)d_in[7],  (const float*)d_in[11]};
    const float* asrc[3] = {(const float*)d_in[4],  (const float*)d_in[8],  (const float*)d_in[12]};
    const float* adst[3] = {(const float*)d_in[5],  (const float*)d_in[9],  (const float*)d_in[13]};
    const float* bias[3] = {(const float*)d_in[6],  (const float*)d_in[10], (const float*)d_in[14]};
    const float* lw = (const float*)d_in[15];
    const float* lb = (const float*)d_in[16];

    const int N_ = in_sizes[0] / HC;      // F_IN == HC == 128
    const int E_ = in_sizes[1] / 2;
    const int ET = E_ + N_;               // edges + self loops
    const int* srcI = ei;
    const int* dstI = ei + E_;

    // workspace carve-up (floats)
    float* ws    = (float*)d_ws;
    float* Xcur  = ws;                           // N*HC
    float* Hbuf  = Xcur  + (size_t)N_ * HC;      // N*HC
    float* Outb  = Hbuf  + (size_t)N_ * HC;      // N*HC
    float* as    = Outb  + (size_t)N_ * HC;      // N*HEADS
    float* ad    = as    + (size_t)N_ * HEADS;   // N*HEADS
    float* mbuf  = ad    + (size_t)N_ * HEADS;   // N*HEADS
    float* den   = mbuf  + (size_t)N_ * HEADS;   // N*HEADS
    float* ev    = den   + (size_t)N_ * HEADS;   // ET*HEADS
    float* pooled= ev    + (size_t)ET * HEADS;   // NGRAPH*HC
    float* cnts  = pooled + NGRAPH * HC;         // NGRAPH

    const int B = 256;
    const int nTiles   = (N_ + ROWS_PER_BLOCK - 1) / ROWS_PER_BLOCK;
    const int gNH      = (N_ * HEADS + B - 1) / B;
    const int gNQ      = (N_ * (HC / 4) + B - 1) / B;     // quad-vectorized N*HC
    const int gE       = (ET + B - 1) / B;
    const size_t totEQ = (size_t)ET * (HC / 4);
    const int gEQ      = (int)((totEQ + B - 1) / B);

    for (int l = 0; l < 3; ++l) {
        const float* Xin = (l == 0) ? x : Xcur;
        gat_gemm_wmma   <<<nTiles, 256, 0, stream>>>(Xin, Wl[l], Hbuf, N_);
        gat_alpha       <<<gNH, B, 0, stream>>>(Hbuf, asrc[l], adst[l], as, ad, N_);
        gat_init        <<<gNQ, B, 0, stream>>>(Outb, mbuf, den, N_);
        gat_edge_max    <<<gE, B, 0, stream>>>(srcI, dstI, E_, N_, as, ad, ev, mbuf);
        gat_edge_exp    <<<gE, B, 0, stream>>>(dstI, E_, N_, mbuf, ev, den);
        gat_edge_scatter<<<gEQ, B, 0, stream>>>(srcI, dstI, E_, N_, Hbuf, ev, den, Outb);
        gat_epilogue    <<<gNQ, B, 0, stream>>>(Outb, bias[l], Xcur, N_, (l < 2) ? 1 : 0);
    }

    pool_init <<<(NGRAPH * HC + B - 1) / B, B, 0, stream>>>(pooled, cnts);
    pool_sum  <<<gNQ, B, 0, stream>>>(Xcur, batch, pooled, cnts, N_);
    final_head<<<1, NGRAPH, 0, stream>>>(pooled, cnts, lw, lb, (float*)d_out);
}